// custom_rbln_paged_attn_decode_11012296147685
// MI455X (gfx1250) — compile-verified
//
#include <hip/hip_runtime.h>
#include <stdint.h>

#define B_ 16
#define H_ 32
#define HKV_ 8
#define G_ 4
#define D_ 128
#define NBLK_ 32
#define KV_ 4096
#define NSPLIT 4
#define WAVES 4
#define NPART (NSPLIT * WAVES)
#define SPLIT_LEN (KV_ / NSPLIT)       // 1024 positions per split
#define WAVE_LEN (SPLIT_LEN / WAVES)   // 256 positions per wave
#define TILES_PER_WAVE (WAVE_LEN / 16) // 16 tiles of 16 positions
#define KSTR 132                       // LDS row stride (floats): conflict-free frag reads
#define VSTR 136
#define PART_STRIDE (G_ + G_ + G_ * D_) // m[4], l[4], O[4][128] = 520 floats

typedef __attribute__((ext_vector_type(2))) float v2f;
typedef __attribute__((ext_vector_type(8))) float v8f;

// -------- main split-KV flash-decode kernel: one wave = 256 KV positions ----
__global__ __launch_bounds__(128, 1) void pa_decode_main(
    const float* __restrict__ q, const float* __restrict__ knew,
    const float* __restrict__ vnew, const float* __restrict__ mask,
    const float* __restrict__ kcache, const float* __restrict__ vcache,
    const int* __restrict__ seq, const int* __restrict__ btab,
    float* __restrict__ ws)
{
  __shared__ __align__(16) float qs[16 * KSTR];             // Q, scaled, rows>=4 zero
  __shared__ __align__(16) float ksm[WAVES][2][16 * KSTR];  // double-buffered K tiles
  __shared__ __align__(16) float vsm[WAVES][2][16 * VSTR];  // double-buffered V tiles

  const int bk    = blockIdx.x;            // b*HKV + kvh
  const int b     = bk / HKV_;
  const int kvh   = bk % HKV_;
  const int split = blockIdx.y;
  const int tid   = threadIdx.x;
  const int wave  = tid >> 5;
  const int lane  = tid & 31;
  const int hl    = lane & 15;             // M-row / N-col index within 16
  const int half  = lane >> 4;             // 0: lanes 0-15, 1: lanes 16-31
  const int sq    = seq[b];
  const float scale = 0.08838834764831845f; // 1/sqrt(128)
  const long newoff = (long)(b * HKV_ + kvh) * D_;

  // Issue one tile's K+V rows as async global->LDS DMA (32 x b128-per-lane).
  auto issue_tile = [&](int tb, int parity) {
    const int pblk = btab[b * NBLK_ + (tb >> 7)];
    const long cbase = (((long)pblk * HKV_ + kvh) * 128 + (tb & 127)) * (long)D_;
    const unsigned kbase = (unsigned)(uintptr_t)(&ksm[wave][parity][0]) + lane * 16u;
    const unsigned vbase = (unsigned)(uintptr_t)(&vsm[wave][parity][0]) + lane * 16u;
#pragma unroll
    for (int r = 0; r < 16; ++r) {
      const int pr = tb + r;
      const float* krow = (pr == sq) ? (knew + newoff) : (kcache + cbase + (long)r * D_);
      const float* vrow = (pr == sq) ? (vnew + newoff) : (vcache + cbase + (long)r * D_);
      const unsigned long long kga = (unsigned long long)(uintptr_t)krow + lane * 16ull;
      const unsigned long long vga = (unsigned long long)(uintptr_t)vrow + lane * 16ull;
      const unsigned kl = kbase + (unsigned)(r * (KSTR * 4));
      const unsigned vl = vbase + (unsigned)(r * (VSTR * 4));
      asm volatile("global_load_async_to_lds_b128 %0, %1, off"
                   :: "v"(kl), "v"(kga) : "memory");
      asm volatile("global_load_async_to_lds_b128 %0, %1, off"
                   :: "v"(vl), "v"(vga) : "memory");
    }
  };

  const int base0 = split * SPLIT_LEN + wave * WAVE_LEN;

  // Prefetch tile 0 while we stage Q.
  if (base0 <= sq) issue_tile(base0, 0);

  // Stage scaled Q into LDS: qs[h][d] (h = head-in-group, rows 4..15 zero)
  for (int i = tid; i < 16 * D_; i += 128) {
    const int h = i >> 7, dc = i & (D_ - 1);
    float val = 0.0f;
    if (h < G_) val = q[((b * H_ + kvh * G_ + h) * D_) + dc] * scale;
    qs[h * KSTR + dc] = val;
  }
  __syncthreads();

  float m_run = -1e30f;
  float l_run = 0.0f;
  v8f zero8 = {};
  v8f oacc[8];
#pragma unroll
  for (int dt = 0; dt < 8; ++dt) oacc[dt] = zero8;

  for (int t = 0; t < TILES_PER_WAVE; ++t) {
    const int tb = base0 + t * 16;
    if (tb > sq) break;                    // uniform within the wave at runtime

    // Issue next tile's DMA, then wait until the current tile's 32 transfers
    // (the older ones; async loads retire in order) are complete.
    const int tb2 = tb + 16;
    const bool inext = (t + 1 < TILES_PER_WAVE) && (tb2 <= sq);
    if (inext) {
      issue_tile(tb2, (t + 1) & 1);
      asm volatile("s_wait_asynccnt 0x20" ::: "memory");
    } else {
      asm volatile("s_wait_asynccnt 0x0" ::: "memory");
    }

    const float* kt = &ksm[wave][t & 1][0];
    const float* vt = &vsm[wave][t & 1][0];

    // ---- QK^T: 32 chained v_wmma_f32_16x16x4_f32 over D=128 ----------------
    // A (16x4 f32): lane hl = position row; vgpr0/1 = d, d+1 (+2 for half=1)
    v8f sc = {};
#pragma unroll
    for (int kk = 0; kk < 32; ++kk) {
      v2f a  = *(const v2f*)(kt + hl * KSTR + 4 * kk + 2 * half);
      v2f bq = *(const v2f*)(&qs[hl * KSTR + 4 * kk + 2 * half]);
      sc = __builtin_amdgcn_wmma_f32_16x16x4_f32(false, a, false, bq,
                                                 (short)0, sc, false, false);
    }

    // ---- online softmax: lane n owns head n%16 at positions tb+j+8*half ----
    float s[8];
    float tmax = -3e30f;
#pragma unroll
    for (int j = 0; j < 8; ++j) {
      const int pj = tb + j + 8 * half;
      float x = sc[j];
      x = (pj <= sq) ? (x + mask[b * KV_ + pj]) : -1e30f;
      s[j] = x;
      tmax = fmaxf(tmax, x);
    }
    tmax = fmaxf(tmax, __shfl_xor(tmax, 16, 32));
    const float mnew = fmaxf(m_run, tmax);
    const float fac = __expf(m_run - mnew);
    float p[8];
    float tsum = 0.0f;
#pragma unroll
    for (int j = 0; j < 8; ++j) { p[j] = __expf(s[j] - mnew); tsum += p[j]; }
    tsum += __shfl_xor(tsum, 16, 32);
    l_run = l_run * fac + tsum;
    m_run = mnew;
#pragma unroll
    for (int dt = 0; dt < 8; ++dt) oacc[dt] *= fac;   // per-lane head rescale

    // ---- build P B-fragments (4 pos-slices of 4) via cross-half swaps ------
    float psw[8];
#pragma unroll
    for (int j = 0; j < 8; ++j) psw[j] = __shfl_xor(p[j], 16, 32);
    v2f bp[4];
    bp[0][0] = half ? psw[2] : p[0];  bp[0][1] = half ? psw[3] : p[1];
    bp[1][0] = half ? psw[6] : p[4];  bp[1][1] = half ? psw[7] : p[5];
    bp[2][0] = half ? p[2]   : psw[0]; bp[2][1] = half ? p[3]   : psw[1];
    bp[3][0] = half ? p[6]   : psw[4]; bp[3][1] = half ? p[7]   : psw[5];

    // ---- O^T += V^T * P : 8 d-tiles x 4 pos-slices = 32 wmma ---------------
#pragma unroll
    for (int dt = 0; dt < 8; ++dt) {
      const int dc = dt * 16 + hl;
#pragma unroll
      for (int sl = 0; sl < 4; ++sl) {
        const int r0 = sl * 4 + 2 * half;
        v2f av;
        av[0] = vt[r0 * VSTR + dc];
        av[1] = vt[(r0 + 1) * VSTR + dc];
        oacc[dt] = __builtin_amdgcn_wmma_f32_16x16x4_f32(false, av, false, bp[sl],
                                                         (short)0, oacc[dt],
                                                         false, false);
      }
    }
  }

  // ---- write per-wave partial: m[4], l[4], O[4][128] -----------------------
  const int part = split * WAVES + wave;
  float* wsb = ws + ((long)bk * NPART + part) * PART_STRIDE;
  if (hl < G_) {
    if (half == 0) { wsb[hl] = m_run; wsb[G_ + hl] = l_run; }
#pragma unroll
    for (int dt = 0; dt < 8; ++dt) {
      float* od = wsb + 2 * G_ + hl * D_ + dt * 16 + 8 * half;
#pragma unroll
      for (int j = 0; j < 8; ++j) od[j] = oacc[dt][j];
    }
  }
}

// -------- split reduction: merge NPART partials per (b, kvh) ----------------
__global__ __launch_bounds__(128, 1) void pa_decode_reduce(
    const float* __restrict__ ws, float* __restrict__ out)
{
  const int bk = blockIdx.x;
  const int b = bk / HKV_, kvh = bk % HKV_;
  const float* base = ws + (long)bk * NPART * PART_STRIDE;
  for (int idx = threadIdx.x; idx < G_ * D_; idx += 128) {
    const int h = idx >> 7, d = idx & (D_ - 1);
    float mstar = -1e30f;
#pragma unroll
    for (int pz = 0; pz < NPART; ++pz)
      mstar = fmaxf(mstar, base[pz * PART_STRIDE + h]);
    float lsum = 0.0f, osum = 0.0f;
#pragma unroll
    for (int pz = 0; pz < NPART; ++pz) {
      const float* pp = base + pz * PART_STRIDE;
      const float w = __expf(pp[h] - mstar);
      lsum += pp[G_ + h] * w;
      osum += pp[2 * G_ + h * D_ + d] * w;
    }
    out[((b * H_ + kvh * G_ + h) * D_) + d] = osum / lsum;
  }
}

extern "C" void kernel_launch(void* const* d_in, const int* in_sizes, int n_in,
                              void* d_out, int out_size, void* d_ws, size_t ws_size,
                              hipStream_t stream) {
  const float* q    = (const float*)d_in[0];
  const float* k    = (const float*)d_in[1];
  const float* v    = (const float*)d_in[2];
  const float* mask = (const float*)d_in[3];
  const float* kc   = (const float*)d_in[4];
  const float* vc   = (const float*)d_in[5];
  const int*   seq  = (const int*)d_in[6];
  const int*   bt   = (const int*)d_in[7];
  float* ws = (float*)d_ws;   // needs 128*16*520*4 = ~4.1 MB

  dim3 g1(B_ * HKV_, NSPLIT);
  pa_decode_main<<<g1, 128, 0, stream>>>(q, k, v, mask, kc, vc, seq, bt, ws);
  pa_decode_reduce<<<B_ * HKV_, 128, 0, stream>>>(ws, (float*)d_out);
}